// TradeFlowGCN_32289564131493
// MI455X (gfx1250) — compile-verified
//
#include <hip/hip_runtime.h>
#include <hip/hip_bf16.h>

typedef __attribute__((ext_vector_type(2))) float v2f;
typedef __attribute__((ext_vector_type(8))) float v8f;

#define HID 64
#define FEDGE 16
#define DECH 64
#define WPB 8   // waves per block (256 threads, wave32)

static __device__ __forceinline__ v8f wmma4(v2f a, v2f b, v8f c) {
  // V_WMMA_F32_16X16X4_F32: D = A(16x4) * B(4x16) + C(16x16), fp32 exact
  return __builtin_amdgcn_wmma_f32_16x16x4_f32(
      /*neg_a=*/false, a, /*neg_b=*/false, b,
      /*c_mod=*/(short)0, c, /*reuse_a=*/false, /*reuse_b=*/false);
}

// ---------------- weight repack: Wp[(k/2)*2N + 2n + (k&1)] = W[k*N + n] ----
// Makes each WMMA B fragment (W[k][n], W[k+1][n]) one aligned 8-byte load.
__global__ void pack_w_kernel(const float* __restrict__ W, float* __restrict__ Wp,
                              int K, int N) {
  int i = blockIdx.x * blockDim.x + threadIdx.x;
  if (i >= K * N) return;
  int k = i / N, n = i - k * N;
  Wp[(k >> 1) * (2 * N) + 2 * n + (k & 1)] = W[i];
}

// ---------------- degree / symmetric norm ----------------
__global__ void deg_init_kernel(float* deg, int n) {
  int i = blockIdx.x * blockDim.x + threadIdx.x;
  if (i < n) deg[i] = 1.0f;  // self-loop contribution
}

__global__ void deg_accum_kernel(const int* __restrict__ dst, float* deg, int ne) {
  int i = blockIdx.x * blockDim.x + threadIdx.x;
  if (i < ne) atomicAdd(&deg[dst[i]], 1.0f);
}

__global__ void deg_finish_kernel(float* deg, int n) {
  int i = blockIdx.x * blockDim.x + threadIdx.x;
  if (i < n) deg[i] = rsqrtf(deg[i]);  // deg >= 1 always
}

// ---------------- node GEMM: T = H @ W  (n x 64) @ (64 x 64) ----------------
// One wave = two 16-row M-tiles (B fragments reused across both tiles).
__global__ void gemm_node_kernel(const float* __restrict__ H,
                                 const float* __restrict__ Wp,
                                 float* __restrict__ T, int n) {
  int wave = blockIdx.x * WPB + (threadIdx.x >> 5);
  int lane = threadIdx.x & 31;
  int mlo  = lane & 15;
  int hi   = lane >> 4;
  int m0   = wave * 32;

  int r0 = m0 + mlo;      if (r0 >= n) r0 = n - 1;  // clamp loads
  int r1 = m0 + 16 + mlo; if (r1 >= n) r1 = n - 1;
  const float* h0 = H + (size_t)r0 * HID;
  const float* h1 = H + (size_t)r1 * HID;

  v8f acc0[4] = {v8f{0.f}, v8f{0.f}, v8f{0.f}, v8f{0.f}};
  v8f acc1[4] = {v8f{0.f}, v8f{0.f}, v8f{0.f}, v8f{0.f}};
#pragma unroll
  for (int ks = 0; ks < 16; ++ks) {
    int k = ks * 4 + 2 * hi;
    v2f a0 = *(const v2f*)(h0 + k);
    v2f a1 = *(const v2f*)(h1 + k);
    const float* wrow = Wp + (k >> 1) * (2 * HID);
#pragma unroll
    for (int nt = 0; nt < 4; ++nt) {
      v2f b = *(const v2f*)(wrow + 2 * (nt * 16 + mlo));
      acc0[nt] = wmma4(a0, b, acc0[nt]);
      acc1[nt] = wmma4(a1, b, acc1[nt]);
    }
  }
#pragma unroll
  for (int nt = 0; nt < 4; ++nt)
#pragma unroll
    for (int v = 0; v < 8; ++v) {
      int ra = m0 + v + 8 * hi;
      int rb = ra + 16;
      if (ra < n) T[(size_t)ra * HID + nt * 16 + mlo] = acc0[nt][v];
      if (rb < n) T[(size_t)rb * HID + nt * 16 + mlo] = acc1[nt][v];
    }
}

// ---------------- self-loop init: agg = T * dinv^2 ----------------
__global__ void scatter_self_kernel(const float* __restrict__ T,
                                    const float* __restrict__ dinv,
                                    float* __restrict__ agg, int n) {
  int i = blockIdx.x * blockDim.x + threadIdx.x;
  if (i < n * HID) {
    float di = dinv[i >> 6];
    agg[i] = T[i] * di * di;
  }
}

// ---------------- edge scatter: agg[dst] += T[src] * dinv[s]*dinv[d] -------
__global__ void scatter_edge_kernel(const float* __restrict__ T,
                                    const float* __restrict__ dinv,
                                    const int* __restrict__ src,
                                    const int* __restrict__ dst,
                                    float* __restrict__ agg, int ne) {
  int e    = blockIdx.x * WPB + (threadIdx.x >> 5);
  int lane = threadIdx.x & 31;
  if (e >= ne) return;
  int s = src[e], d = dst[e];
  float nrm = dinv[s] * dinv[d];
  const float* ts = T + (size_t)s * HID;
  float* ad = agg + (size_t)d * HID;
  atomicAdd(&ad[lane],      ts[lane]      * nrm);
  atomicAdd(&ad[lane + 32], ts[lane + 32] * nrm);
}

// ---------------- bias + layernorm + relu (one wave per node) --------------
__global__ void ln_relu_kernel(const float* __restrict__ agg,
                               const float* __restrict__ bias,
                               const float* __restrict__ g,
                               const float* __restrict__ b,
                               float* __restrict__ Hout, int n) {
  int node = blockIdx.x * WPB + (threadIdx.x >> 5);
  int lane = threadIdx.x & 31;
  if (node >= n) return;
  const float* a = agg + (size_t)node * HID;
  float v0 = a[lane]      + bias[lane];
  float v1 = a[lane + 32] + bias[lane + 32];
  float s = v0 + v1;
#pragma unroll
  for (int off = 16; off > 0; off >>= 1) s += __shfl_xor(s, off, 32);
  float mu = s * (1.0f / HID);
  float d0 = v0 - mu, d1 = v1 - mu;
  float q = d0 * d0 + d1 * d1;
#pragma unroll
  for (int off = 16; off > 0; off >>= 1) q += __shfl_xor(q, off, 32);
  float rstd = rsqrtf(q * (1.0f / HID) + 1e-5f);
  float o0 = d0 * rstd * g[lane]      + b[lane];
  float o1 = d1 * rstd * g[lane + 32] + b[lane + 32];
  float* ho = Hout + (size_t)node * HID;
  ho[lane]      = o0 > 0.f ? o0 : 0.f;
  ho[lane + 32] = o1 > 0.f ? o1 : 0.f;
}

// ---------------- fused edge decoder MLP (16 edges per wave) ---------------
// z = [h[src] | h[dst] | ea] (16x144) -> relu(z@W1+b1) (16x64)
//   -> relu(@W2+b2) (16x32) -> @w3+b3 (16)
__global__ void decoder_kernel(const float* __restrict__ H,
                               const int* __restrict__ src,
                               const int* __restrict__ dst,
                               const float* __restrict__ EA,
                               const float* __restrict__ W1p, const float* __restrict__ B1,
                               const float* __restrict__ W2p, const float* __restrict__ B2,
                               const float* __restrict__ W3, const float* __restrict__ B3,
                               float* __restrict__ out, int ne) {
  __shared__ __align__(16) float z1s[WPB][16 * DECH];       // 32 KB
  __shared__ __align__(16) float z2s[WPB][16 * (DECH / 2)]; // 16 KB

  int wv   = threadIdx.x >> 5;
  int lane = threadIdx.x & 31;
  int mlo  = lane & 15;
  int hi   = lane >> 4;
  int e0   = (blockIdx.x * WPB + wv) * 16;

  int e = e0 + mlo;
  if (e >= ne) e = ne - 1;  // clamp; final store predicated
  int s = src[e], d = dst[e];
  const float* hs = H + (size_t)s * HID;
  const float* hd = H + (size_t)d * HID;
  const float* ea = EA + (size_t)e * FEDGE;

  // ---- layer 1: K = 144 (h[src] 0..63 | h[dst] 64..127 | ea 128..143) ----
  v8f acc[4] = {v8f{0.f}, v8f{0.f}, v8f{0.f}, v8f{0.f}};
#pragma unroll
  for (int ks = 0; ks < 36; ++ks) {
    int k = ks * 4 + 2 * hi;  // even pair, never crosses the 64/128 boundaries
    v2f a;
    if (k < 64)        a = *(const v2f*)(hs + k);
    else if (k < 128)  a = *(const v2f*)(hd + (k - 64));
    else               a = *(const v2f*)(ea + (k - 128));
    const float* wrow = W1p + (k >> 1) * (2 * DECH);
#pragma unroll
    for (int nt = 0; nt < 4; ++nt) {
      v2f b = *(const v2f*)(wrow + 2 * (nt * 16 + mlo));
      acc[nt] = wmma4(a, b, acc[nt]);
    }
  }
  // bias + relu, transpose through LDS (C layout -> A layout)
#pragma unroll
  for (int nt = 0; nt < 4; ++nt) {
    int col = nt * 16 + mlo;
    float bb = B1[col];
#pragma unroll
    for (int v = 0; v < 8; ++v) {
      float z = acc[nt][v] + bb;
      z1s[wv][(v + 8 * hi) * DECH + col] = z > 0.f ? z : 0.f;
    }
  }
  __syncthreads();

  // ---- layer 2: (16x64) @ (64x32) ----
  v8f acc2[2] = {v8f{0.f}, v8f{0.f}};
#pragma unroll
  for (int ks = 0; ks < 16; ++ks) {
    int k = ks * 4 + 2 * hi;
    v2f a = *(const v2f*)(&z1s[wv][mlo * DECH + k]);
    const float* wrow = W2p + (k >> 1) * (DECH);  // 2N = 64
#pragma unroll
    for (int nt = 0; nt < 2; ++nt) {
      v2f b = *(const v2f*)(wrow + 2 * (nt * 16 + mlo));
      acc2[nt] = wmma4(a, b, acc2[nt]);
    }
  }
#pragma unroll
  for (int nt = 0; nt < 2; ++nt) {
    int col = nt * 16 + mlo;
    float bb = B2[col];
#pragma unroll
    for (int v = 0; v < 8; ++v) {
      float z = acc2[nt][v] + bb;
      z2s[wv][(v + 8 * hi) * (DECH / 2) + col] = z > 0.f ? z : 0.f;
    }
  }
  __syncthreads();

  // ---- layer 3: dot-32 per edge (lanes 0..15) ----
  if (lane < 16) {
    int eo = e0 + lane;
    if (eo < ne) {
      float sum = B3[0];
      const float* zr = &z2s[wv][lane * (DECH / 2)];
#pragma unroll
      for (int j = 0; j < DECH / 2; ++j) sum += zr[j] * W3[j];
      out[eo] = sum;
    }
  }
}

extern "C" void kernel_launch(void* const* d_in, const int* in_sizes, int n_in,
                              void* d_out, int out_size, void* d_ws, size_t ws_size,
                              hipStream_t stream) {
  (void)n_in; (void)out_size; (void)ws_size;
  const float* x      = (const float*)d_in[0];
  const int*   ei     = (const int*)  d_in[1];
  const float* eattr  = (const float*)d_in[2];
  const float* conv_w = (const float*)d_in[3];
  const float* conv_b = (const float*)d_in[4];
  const float* ln_g   = (const float*)d_in[5];
  const float* ln_b   = (const float*)d_in[6];
  const float* w1     = (const float*)d_in[7];
  const float* b1     = (const float*)d_in[8];
  const float* w2     = (const float*)d_in[9];
  const float* b2     = (const float*)d_in[10];
  const float* w3     = (const float*)d_in[11];
  const float* b3     = (const float*)d_in[12];
  float* out = (float*)d_out;

  int n  = in_sizes[0] / HID;  // 50000
  int ne = in_sizes[1] / 2;    // 800000
  const int* src = ei;
  const int* dst = ei + ne;

  // workspace carve-out (256B aligned)
  char* ws = (char*)d_ws;
  size_t off = 0;
  auto carve = [&](size_t bytes) -> float* {
    float* p = (float*)(ws + off);
    off += (bytes + 255) & ~(size_t)255;
    return p;
  };
  float* dinv  = carve((size_t)n * sizeof(float));
  float* hbuf  = carve((size_t)n * HID * sizeof(float));
  float* tbuf  = carve((size_t)n * HID * sizeof(float));
  float* abuf  = carve((size_t)n * HID * sizeof(float));
  float* cwp   = carve((size_t)3 * HID * HID * sizeof(float));          // packed conv_w
  float* w1p   = carve((size_t)(2 * HID + FEDGE) * DECH * sizeof(float)); // packed dec_w1
  float* w2p   = carve((size_t)DECH * (DECH / 2) * sizeof(float));       // packed dec_w2

  const int TB = 256;

  // pack weights into paired-row layout (one b64 per WMMA B fragment)
  for (int l = 0; l < 3; ++l)
    pack_w_kernel<<<(HID * HID + TB - 1) / TB, TB, 0, stream>>>(
        conv_w + (size_t)l * HID * HID, cwp + (size_t)l * HID * HID, HID, HID);
  pack_w_kernel<<<((2 * HID + FEDGE) * DECH + TB - 1) / TB, TB, 0, stream>>>(
      w1, w1p, 2 * HID + FEDGE, DECH);
  pack_w_kernel<<<(DECH * (DECH / 2) + TB - 1) / TB, TB, 0, stream>>>(
      w2, w2p, DECH, DECH / 2);

  // degree / norm
  deg_init_kernel<<<(n + TB - 1) / TB, TB, 0, stream>>>(dinv, n);
  deg_accum_kernel<<<(ne + TB - 1) / TB, TB, 0, stream>>>(dst, dinv, ne);
  deg_finish_kernel<<<(n + TB - 1) / TB, TB, 0, stream>>>(dinv, n);

  int mtiles32 = (n + 31) / 32;
  for (int l = 0; l < 3; ++l) {
    const float* hin = (l == 0) ? x : hbuf;
    gemm_node_kernel<<<(mtiles32 + WPB - 1) / WPB, TB, 0, stream>>>(
        hin, cwp + (size_t)l * HID * HID, tbuf, n);
    scatter_self_kernel<<<((size_t)n * HID + TB - 1) / TB, TB, 0, stream>>>(tbuf, dinv, abuf, n);
    scatter_edge_kernel<<<(ne + WPB - 1) / WPB, TB, 0, stream>>>(tbuf, dinv, src, dst, abuf, ne);
    ln_relu_kernel<<<(n + WPB - 1) / WPB, TB, 0, stream>>>(
        abuf, conv_b + (size_t)l * HID, ln_g + (size_t)l * HID, ln_b + (size_t)l * HID, hbuf, n);
  }

  int etiles = (ne + 15) / 16;
  decoder_kernel<<<(etiles + WPB - 1) / WPB, TB, 0, stream>>>(
      hbuf, src, dst, eattr, w1p, b1, w2p, b2, w3, b3, out, ne);
}